// MULTI_GCN_43542378447163
// MI455X (gfx1250) — compile-verified
//
#include <hip/hip_runtime.h>
#include <hip/hip_bf16.h>
#include <math.h>

typedef __attribute__((ext_vector_type(2))) float v2f;
typedef __attribute__((ext_vector_type(8))) float v8f;

#define HFEAT 128   // hidden / input feature width (K of every GEMM)

// ---------------------------------------------------------------------------
// Degree / norm kernels
// ---------------------------------------------------------------------------
__global__ void k_deg_init(int* __restrict__ deg, int n) {
    int i = blockIdx.x * blockDim.x + threadIdx.x;
    if (i < n) deg[i] = 1;                       // self-loop contributes 1
}

__global__ void k_deg_count(const int* __restrict__ dst, int* __restrict__ deg, int E) {
    int e = blockIdx.x * blockDim.x + threadIdx.x;
    if (e < E) atomicAdd(&deg[dst[e]], 1);
}

__global__ void k_dinv(const int* __restrict__ deg, float* __restrict__ dinv, int n) {
    int i = blockIdx.x * blockDim.x + threadIdx.x;
    if (i < n) dinv[i] = rsqrtf((float)deg[i]);  // deg >= 1 always
}

// ---------------------------------------------------------------------------
// fp32 WMMA GEMM:  Y[M,Fout] = X[M,128] @ W[128,Fout]
// One wave computes a 16-row stripe across all NT 16-col tiles.
//
// A-frag (16x4 f32): lane<16 -> M=lane, K={k,k+1}; lane>=16 -> K={k+2,k+3}
//   -> one contiguous 8B global load per lane per k-step.
// B-frag (4x16 f32): VGPR0 = row k+ksel, VGPR1 = row k+ksel+1 (ksel=2*(lane>>4))
//   -> W is pre-SWIZZLED in LDS so both elements are one contiguous float2:
//      Wsw[((k/4)*2 + pair)*NPAD + n] = { W[4*(k/4)+2*pair+0][n],
//                                          W[4*(k/4)+2*pair+1][n] }
//   -> single ds_load_b64 into an even-aligned VGPR pair, no re-pack movs.
// D layout: vgpr v, lane l -> row m0 + v + 8*(l>>4), col n0 + (l&15)
// ---------------------------------------------------------------------------
template <int NT>
__global__ __launch_bounds__(256) void k_gemm_wmma(const float* __restrict__ X,
                                                   const float* __restrict__ W,
                                                   float* __restrict__ Y,
                                                   int M, int Fout) {
    constexpr int NPAD = NT * 16;
    __shared__ v2f Wsw[(HFEAT / 2) * NPAD];      // swizzled weights, 8B granules

    // cooperative swizzled stage of W into LDS (zero-pad cols for Fout < NPAD)
    for (int idx = threadIdx.x; idx < HFEAT * NPAD; idx += blockDim.x) {
        int k = idx / NPAD, n = idx % NPAD;
        int kb = k >> 2;              // k-block of 4
        int p  = (k >> 1) & 1;        // which lane-half pair
        int j  = k & 1;               // element within the float2
        float v = (n < Fout) ? W[k * Fout + n] : 0.0f;
        Wsw[(kb * 2 + p) * NPAD + n][j] = v;
    }
    __syncthreads();

    const int wave = blockIdx.x * (blockDim.x >> 5) + (threadIdx.x >> 5);
    const int lane = threadIdx.x & 31;
    const int m0 = wave * 16;
    if (m0 >= M) return;                          // wave-uniform: EXEC stays all-1s

    const int arow = m0 + (lane & 15);
    const int pair = lane >> 4;                   // 0 or 1
    const int ksel = 2 * pair;
    const int ncol = lane & 15;

    v8f acc[NT] = {};

    const float* __restrict__ xrow = X + (size_t)arow * HFEAT;
    const v2f* __restrict__ brow = Wsw + pair * NPAD + ncol;

    #pragma unroll 4
    for (int k = 0; k < HFEAT; k += 4) {
        v2f a = *(const v2f*)(xrow + k + ksel);
        const v2f* bk = brow + (k >> 2) * (2 * NPAD);
        #pragma unroll
        for (int t = 0; t < NT; ++t) {
            v2f b = bk[t * 16];
            acc[t] = __builtin_amdgcn_wmma_f32_16x16x4_f32(
                false, a, false, b, (short)0, acc[t], false, false);
        }
    }

    const int rbase = m0 + 8 * pair;
    #pragma unroll
    for (int t = 0; t < NT; ++t) {
        int col = t * 16 + ncol;
        if (col < Fout) {
            #pragma unroll
            for (int v = 0; v < 8; ++v)
                Y[(size_t)(rbase + v) * Fout + col] = acc[t][v];
        }
    }
}

// ---------------------------------------------------------------------------
// agg[i,:] = h[i,:] * dinv[i]^2   (self-loop term; also initializes agg)
// ---------------------------------------------------------------------------
__global__ void k_selfloop_init(const float* __restrict__ h, const float* __restrict__ dinv,
                                float* __restrict__ agg, int n, int F) {
    int idx = blockIdx.x * blockDim.x + threadIdx.x;
    if (idx < n * F) {
        int i = idx / F;
        float d = dinv[i];
        agg[idx] = h[idx] * d * d;
    }
}

// ---------------------------------------------------------------------------
// Edge scatter: one wave per edge, lane covers 4 consecutive features.
// agg[dst] += h[src] * dinv[src]*dinv[dst]
// ---------------------------------------------------------------------------
__global__ __launch_bounds__(256) void k_scatter(const float* __restrict__ h,
                                                 const int* __restrict__ src,
                                                 const int* __restrict__ dst,
                                                 const float* __restrict__ dinv,
                                                 float* __restrict__ agg,
                                                 int E, int F) {
    int e = blockIdx.x * (blockDim.x >> 5) + (threadIdx.x >> 5);
    int lane = threadIdx.x & 31;
    if (e >= E) return;
    int s = src[e], d = dst[e];
    float w = dinv[s] * dinv[d];
    int f0 = lane * 4;
    if (f0 < F) {
        const float4 hv = *(const float4*)(h + (size_t)s * F + f0);
        float* ap = agg + (size_t)d * F + f0;
        atomicAdd(ap + 0, hv.x * w);
        atomicAdd(ap + 1, hv.y * w);
        atomicAdd(ap + 2, hv.z * w);
        atomicAdd(ap + 3, hv.w * w);
    }
}

// ---------------------------------------------------------------------------
// BatchNorm stats: one block per column; deterministic LDS tree reduction.
// Produces fused affine: y = x*scale + shift  where scale = gamma*rstd,
// shift = beta - mu*gamma*rstd.
// ---------------------------------------------------------------------------
__global__ __launch_bounds__(256) void k_bn_stats(const float* __restrict__ x, int n, int F,
                                                  const float* __restrict__ gamma,
                                                  const float* __restrict__ beta,
                                                  float* __restrict__ scale,
                                                  float* __restrict__ shift) {
    const int c = blockIdx.x;
    float s = 0.0f, ss = 0.0f;
    for (int r = threadIdx.x; r < n; r += blockDim.x) {
        float v = x[(size_t)r * F + c];
        s += v;
        ss += v * v;
    }
    __shared__ float sh1[256], sh2[256];
    sh1[threadIdx.x] = s;
    sh2[threadIdx.x] = ss;
    __syncthreads();
    for (int off = 128; off > 0; off >>= 1) {
        if (threadIdx.x < off) {
            sh1[threadIdx.x] += sh1[threadIdx.x + off];
            sh2[threadIdx.x] += sh2[threadIdx.x + off];
        }
        __syncthreads();
    }
    if (threadIdx.x == 0) {
        float inv_n = 1.0f / (float)n;
        float mu = sh1[0] * inv_n;
        float var = sh2[0] * inv_n - mu * mu;        // biased variance
        float rstd = rsqrtf(var + 1e-5f);
        float g = gamma[c] * rstd;
        scale[c] = g;
        shift[c] = beta[c] - mu * g;
    }
}

__global__ void k_bn_relu(float* __restrict__ x, int n, int F,
                          const float* __restrict__ scale, const float* __restrict__ shift) {
    int idx = blockIdx.x * blockDim.x + threadIdx.x;
    if (idx < n * F) {
        int c = idx % F;
        float v = x[idx] * scale[c] + shift[c];
        x[idx] = v > 0.0f ? v : 0.0f;
    }
}

// ---------------------------------------------------------------------------
// out = log_softmax(z + b3) along C columns; one wave per row (C <= 64).
// ---------------------------------------------------------------------------
__global__ __launch_bounds__(256) void k_logsoftmax(const float* __restrict__ z,
                                                    const float* __restrict__ bias,
                                                    float* __restrict__ out, int n, int C) {
    int row = blockIdx.x * (blockDim.x >> 5) + (threadIdx.x >> 5);
    int lane = threadIdx.x & 31;
    if (row >= n) return;
    const float* zr = z + (size_t)row * C;
    const float NEG = -3.4e38f;
    float v0 = (lane < C) ? zr[lane] + bias[lane] : NEG;
    float v1 = (lane + 32 < C) ? zr[lane + 32] + bias[lane + 32] : NEG;
    float m = fmaxf(v0, v1);
    #pragma unroll
    for (int off = 16; off > 0; off >>= 1)
        m = fmaxf(m, __shfl_xor(m, off, 32));
    float e = ((lane < C) ? expf(v0 - m) : 0.0f) +
              ((lane + 32 < C) ? expf(v1 - m) : 0.0f);
    #pragma unroll
    for (int off = 16; off > 0; off >>= 1)
        e += __shfl_xor(e, off, 32);
    float lse = m + logf(e);
    float* orow = out + (size_t)row * C;
    if (lane < C) orow[lane] = v0 - lse;
    if (lane + 32 < C) orow[lane + 32] = v1 - lse;
}

// ---------------------------------------------------------------------------
// Host orchestration
// ---------------------------------------------------------------------------
extern "C" void kernel_launch(void* const* d_in, const int* in_sizes, int n_in,
                              void* d_out, int out_size, void* d_ws, size_t ws_size,
                              hipStream_t stream) {
    const float* x      = (const float*)d_in[0];
    const int*   ei     = (const int*)d_in[1];
    const float* W1     = (const float*)d_in[2];
    const float* gamma1 = (const float*)d_in[4];
    const float* beta1  = (const float*)d_in[5];
    const float* W2     = (const float*)d_in[6];
    const float* gamma2 = (const float*)d_in[8];
    const float* beta2  = (const float*)d_in[9];
    const float* W3     = (const float*)d_in[10];
    const float* b3     = (const float*)d_in[11];
    // b1/b2 cancel exactly through BatchNorm (constant mean shift) -> skipped.

    const int N = in_sizes[0] / HFEAT;     // 50000
    const int E = in_sizes[1] / 2;         // 600000
    const int C = in_sizes[10] / HFEAT;    // 40
    const int* src = ei;
    const int* dst = ei + E;

    // workspace carve-up
    float* bufA = (float*)d_ws;                        // N*128 (GEMM out)
    float* bufB = bufA + (size_t)N * HFEAT;            // N*128 (scatter out / next input)
    int*   deg  = (int*)(bufB + (size_t)N * HFEAT);    // N
    float* dinv = (float*)(deg + N);                   // N
    float* scl  = dinv + N;                            // 128
    float* shf  = scl + HFEAT;                         // 128

    const int TPB = 256;
    const int nEW = (N * HFEAT + TPB - 1) / TPB;       // elementwise blocks [N,128]
    const int nEWc = (N * C + TPB - 1) / TPB;          // elementwise blocks [N,C]
    const int gemmBlocks = ((N / 16) + 7) / 8;         // 8 waves/block, 16 rows/wave
    const int scatBlocks = (E + 7) / 8;                // 8 edges/block
    const int rowBlocks  = (N + 7) / 8;                // 8 rows/block (softmax)

    // ---- normalization coefficients ----
    k_deg_init<<<(N + TPB - 1) / TPB, TPB, 0, stream>>>(deg, N);
    k_deg_count<<<(E + TPB - 1) / TPB, TPB, 0, stream>>>(dst, deg, E);
    k_dinv<<<(N + TPB - 1) / TPB, TPB, 0, stream>>>(deg, dinv, N);

    // ---- layer 1: GCNConv(128->128) + BN + ReLU ----
    k_gemm_wmma<8><<<gemmBlocks, TPB, 0, stream>>>(x, W1, bufA, N, HFEAT);
    k_selfloop_init<<<nEW, TPB, 0, stream>>>(bufA, dinv, bufB, N, HFEAT);
    k_scatter<<<scatBlocks, TPB, 0, stream>>>(bufA, src, dst, dinv, bufB, E, HFEAT);
    k_bn_stats<<<HFEAT, TPB, 0, stream>>>(bufB, N, HFEAT, gamma1, beta1, scl, shf);
    k_bn_relu<<<nEW, TPB, 0, stream>>>(bufB, N, HFEAT, scl, shf);

    // ---- layer 2: GCNConv(128->128) + BN + ReLU ----
    k_gemm_wmma<8><<<gemmBlocks, TPB, 0, stream>>>(bufB, W2, bufA, N, HFEAT);
    k_selfloop_init<<<nEW, TPB, 0, stream>>>(bufA, dinv, bufB, N, HFEAT);
    k_scatter<<<scatBlocks, TPB, 0, stream>>>(bufA, src, dst, dinv, bufB, E, HFEAT);
    k_bn_stats<<<HFEAT, TPB, 0, stream>>>(bufB, N, HFEAT, gamma2, beta2, scl, shf);
    k_bn_relu<<<nEW, TPB, 0, stream>>>(bufB, N, HFEAT, scl, shf);

    // ---- layer 3: GCNConv(128->40) + bias + log_softmax ----
    k_gemm_wmma<3><<<gemmBlocks, TPB, 0, stream>>>(bufB, W3, bufA, N, C);
    k_selfloop_init<<<nEWc, TPB, 0, stream>>>(bufA, dinv, bufB, N, C);
    k_scatter<<<scatBlocks, TPB, 0, stream>>>(bufA, src, dst, dinv, bufB, E, C);
    k_logsoftmax<<<rowBlocks, TPB, 0, stream>>>(bufB, b3, (float*)d_out, N, C);
}